// GraphConvolution_22204980920810
// MI455X (gfx1250) — compile-verified
//
#include <hip/hip_runtime.h>
#include <hip/hip_bf16.h>

typedef __attribute__((ext_vector_type(2))) float v2f;
typedef __attribute__((ext_vector_type(8))) float v8f;
typedef __attribute__((ext_vector_type(4))) unsigned int u32x4;
typedef __attribute__((ext_vector_type(4))) int i32x4;
typedef __attribute__((ext_vector_type(8))) int i32x8;

#define D 128           // feature dim
#define U 128           // output units
#define LDS_STRIDE 136  // 128 + 8 floats pad -> K and K+2 rows hit disjoint LDS banks

// ---------------------------------------------------------------------------
// Kernel 1: zero the accumulators (harness poisons d_ws with 0xAA).
// ---------------------------------------------------------------------------
__global__ void zero_ws_kernel(float* __restrict__ p, long long n) {
    long long i = (long long)blockIdx.x * blockDim.x + threadIdx.x;
    long long stride = (long long)gridDim.x * blockDim.x;
    for (; i < n; i += stride) p[i] = 0.0f;
}

// ---------------------------------------------------------------------------
// Kernel 2: edge scatter. One wave32 per edge; each lane handles 4 floats.
//   num[src] += w * feat[dst]   (fp32 L2 atomics, no-return -> STOREcnt)
//   den[src] += w
// ---------------------------------------------------------------------------
__global__ void edge_scatter_kernel(const float* __restrict__ feat,
                                    const int*   __restrict__ esrc,
                                    const int*   __restrict__ edst,
                                    const float* __restrict__ ew,
                                    float* __restrict__ num,
                                    float* __restrict__ den,
                                    int E) {
    const int lane = threadIdx.x & 31;
    const int e = (int)(((long long)blockIdx.x * blockDim.x + threadIdx.x) >> 5);
    if (e >= E) return;

    const int   s = esrc[e];
    const int   d = edst[e];
    const float w = ew[e];

    const float4* srcrow = (const float4*)(feat + (size_t)d * D);
    float4 v = srcrow[lane];                      // 32 lanes x 16B = 512B row

    float* dst = num + (size_t)s * D + lane * 4;
    atomicAdd(dst + 0, v.x * w);
    atomicAdd(dst + 1, v.y * w);
    atomicAdd(dst + 2, v.z * w);
    atomicAdd(dst + 3, v.w * w);
    if (lane == 0) atomicAdd(den + s, w);
}

// ---------------------------------------------------------------------------
// TDM staging of W (128x128 f32) global -> LDS with row padding.
// D# per CDNA5 ISA §8.3/8.4:
//   group0: count=1, lds_addr, global_addr[56:0], type=2
//   group1: data_size=2 (4B), pad_enable, pad_interval=6 (128 DWORDs = 1 row),
//           pad_amount=7 (8 DWORDs)  -> LDS row stride = 136 floats,
//           tensor_dim0=128, tensor_dim1=128, tile_dim0=128, tile_dim1=128,
//           tensor_dim0_stride=128 (data_size units)
// ---------------------------------------------------------------------------
#if __has_builtin(__builtin_amdgcn_tensor_load_to_lds)
#define USE_TDM 1
__device__ inline void tdm_stage_W(const float* Wg, unsigned lds_byte_off) {
    unsigned long long ga = (unsigned long long)(uintptr_t)Wg;
    u32x4 g0;
    g0.x = 1u;                                        // count=1, user descriptor
    g0.y = lds_byte_off;                              // lds_addr (bytes)
    g0.z = (unsigned)(ga & 0xFFFFFFFFu);              // global_addr[31:0]
    g0.w = (unsigned)((ga >> 32) & 0x01FFFFFFu)       // global_addr[56:32]
         | (2u << 30);                                // type=2 ("image")
    i32x8 g1;
    g1[0] = (2 << 16)                                 // data_size = 4 bytes
          | (1 << 20)                                 // pad_enable
          | (6 << 22)                                 // pad_interval: 128 DWORDs
          | (7 << 25);                                // pad_amount: 8 DWORDs
    g1[1] = (U << 16);                                // tensor_dim0 = 128 (bits 63:48)
    g1[2] = (U << 16);                                // tensor_dim1 = 128 (bits 95:80)
    g1[3] = (U << 16);                                // tile_dim0   = 128 (bits 127:112)
    g1[4] = U;                                        // tile_dim1   = 128 (bits 143:128)
    g1[5] = U;                                        // tensor_dim0_stride = 128 (low 32)
    g1[6] = 0;                                        // stride0 hi / stride1 lo
    g1[7] = 0;
    i32x4 gz4 = {0, 0, 0, 0};
#if __clang_major__ >= 23
    i32x8 gz8 = {0, 0, 0, 0, 0, 0, 0, 0};
    __builtin_amdgcn_tensor_load_to_lds(g0, g1, gz4, gz4, gz8, 0);
#else
    __builtin_amdgcn_tensor_load_to_lds(g0, g1, gz4, gz4, 0);
#endif
}
#else
#define USE_TDM 0
#endif

// ---------------------------------------------------------------------------
// Kernel 3: out = relu((num * (1/max(den,1e-12))) @ W + b)
// One wave per 16-row slab; 8 column tiles of 16; V_WMMA_F32_16X16X4_F32.
// W staged in LDS via TDM (padded stride -> conflict-free ds_load_2addr_b32).
// A fragments use a depth-2 rolling prefetch (4 VGPRs) so WMMAs never sit on
// s_wait_loadcnt 0, without a spill-prone register array.
//
// fp32 WMMA fragment layouts (ISA 7.12.2, wave32):
//   A (16x4):  lane L holds A[L%16][2*(L/16)+v] in VGPR v (v=0,1)
//   B (4x16):  lane L holds B[2*(L/16)+v][L%16] in VGPR v
//   C/D(16x16):lane L, VGPR r holds C[r + 8*(L/16)][L%16]
// ---------------------------------------------------------------------------
__global__ void __launch_bounds__(256)
gemm_bias_relu_kernel(const float* __restrict__ num,
                      const float* __restrict__ den,
                      const float* __restrict__ W,
                      const float* __restrict__ bias,
                      float* __restrict__ out,
                      int N) {
    __shared__ float ldsW[D * LDS_STRIDE];            // 69,632 B

    const int lane = threadIdx.x & 31;
    const int wave = threadIdx.x >> 5;

    // ---- stage W into LDS (whole block participates in the barrier) ----
#if USE_TDM
    if (wave == 0) {
        tdm_stage_W(W, (unsigned)(uintptr_t)&ldsW[0]);
        __builtin_amdgcn_s_wait_tensorcnt(0);
    }
#else
    {
        const int t = threadIdx.x;                    // 256 threads, 128 rows
        const int r = t >> 1;                         // row 0..127
        const int h = t & 1;                          // half-row
        const float4* src = (const float4*)(W + (size_t)r * U + h * 64);
        float4* dst = (float4*)(&ldsW[r * LDS_STRIDE + h * 64]);
#pragma unroll
        for (int i = 0; i < 16; ++i) dst[i] = src[i];
    }
#endif
    __syncthreads();

    const int row0 = (blockIdx.x * 8 + wave) * 16;    // uniform per wave
    if (row0 >= N) return;                            // uniform exit after barrier

    const int m16  = lane & 15;
    const int half = lane >> 4;

    // Per-lane A row (row0 + m16); fold 1/den into A-fragment use.
    int rowA = row0 + m16;
    if (rowA >= N) rowA = N - 1;                      // clamp (N%16==0 so never hit)
    const float  dv   = den[rowA];
    const float  rden = 1.0f / fmaxf(dv, 1e-12f);
    const float* arow = num + (size_t)rowA * D + 2 * half;

    v8f acc[8];
#pragma unroll
    for (int t = 0; t < 8; ++t) acc[t] = (v8f){0.f,0.f,0.f,0.f,0.f,0.f,0.f,0.f};

    // Depth-2 rolling prefetch of A fragments (b64 each).
    v2f abuf[2];
    abuf[0] = *(const v2f*)(arow);                    // i = 0
    abuf[1] = *(const v2f*)(arow + 4);                // i = 1

#pragma unroll 4
    for (int i = 0; i < 32; ++i) {
        const v2f a_cur = abuf[i & 1];
        if (i + 2 < 32)                               // static per unrolled iter
            abuf[i & 1] = *(const v2f*)(arow + (i + 2) * 4);

        v2f a;
        a.x = a_cur.x * rden;
        a.y = a_cur.y * rden;

        const int ka = i * 4 + 2 * half;
        const float* brow0 = &ldsW[ka * LDS_STRIDE];
        const float* brow1 = &ldsW[(ka + 1) * LDS_STRIDE];
#pragma unroll
        for (int t = 0; t < 8; ++t) {
            const int col = t * 16 + m16;
            v2f bfrag;
            bfrag.x = brow0[col];
            bfrag.y = brow1[col];
            acc[t] = __builtin_amdgcn_wmma_f32_16x16x4_f32(
                /*neg_a=*/false, a, /*neg_b=*/false, bfrag,
                /*c_mod=*/(short)0, acc[t],
                /*reuse_a=*/false, /*reuse_b=*/false);
        }
    }

    // Fused bias + ReLU store. D-matrix: VGPR r -> row (r + 8*half), col m16.
#pragma unroll
    for (int t = 0; t < 8; ++t) {
        const int col = t * 16 + m16;
        const float bb = bias[col];
#pragma unroll
        for (int r = 0; r < 8; ++r) {
            const int row = row0 + r + 8 * half;
            if (row < N) {
                float v = acc[t][r] + bb;
                out[(size_t)row * U + col] = fmaxf(v, 0.0f);
            }
        }
    }
}

// ---------------------------------------------------------------------------
// Launch: zero ws -> scatter edges -> WMMA GEMM (+bias+relu)
// Workspace layout: num[N*128] floats, then den[N] floats (~25.8 MB).
// ---------------------------------------------------------------------------
extern "C" void kernel_launch(void* const* d_in, const int* in_sizes, int n_in,
                              void* d_out, int out_size, void* d_ws, size_t ws_size,
                              hipStream_t stream) {
    const float* feat = (const float*)d_in[0];   // [N, 128]
    const int*   esrc = (const int*)  d_in[1];   // [E]
    const int*   edst = (const int*)  d_in[2];   // [E]
    const float* ew   = (const float*)d_in[3];   // [E]
    const float* W    = (const float*)d_in[4];   // [128, 128]
    const float* bias = (const float*)d_in[5];   // [128]
    float*       out  = (float*)d_out;           // [N, 128]

    const int N = in_sizes[0] / D;               // 50000
    const int E = in_sizes[1];                   // 1,600,000

    float* num = (float*)d_ws;                   // N*128 floats
    float* den = num + (size_t)N * D;            // N floats
    const long long zcount = (long long)N * D + N;

    // 1) zero accumulators
    {
        int threads = 256;
        int blocks = (int)((zcount + threads - 1) / threads);
        if (blocks > 65535 * 8) blocks = 65535 * 8;
        zero_ws_kernel<<<blocks, threads, 0, stream>>>(num, zcount);
    }

    // 2) edge scatter: one wave32 per edge
    {
        long long total_threads = (long long)E * 32;
        int threads = 256;
        int blocks = (int)((total_threads + threads - 1) / threads);
        edge_scatter_kernel<<<blocks, threads, 0, stream>>>(
            feat, esrc, edst, ew, num, den, E);
    }

    // 3) normalize + GEMM + bias + relu (WMMA fp32, W staged via TDM)
    {
        int threads = 256;                       // 8 waves -> 128 rows per block
        int blocks = (N + 127) / 128;
        gemm_bias_relu_kernel<<<blocks, threads, 0, stream>>>(
            num, den, W, bias, out, N);
    }
}